// SupConLoss_32710470926983
// MI455X (gfx1250) — compile-verified
//
#include <hip/hip_runtime.h>
#include <hip/hip_bf16.h>
#include <math.h>

typedef float v2f __attribute__((ext_vector_type(2)));
typedef float v8f __attribute__((ext_vector_type(8)));

#define N_ROWS 8192
#define DIM 256
#define INV_TEMP (1.0f / 0.07f)
#define LDS_STRIDE 260        // 256 + 4 pad: float4-aligned (1040B) and bank-conflict-free
#define SLICES 4              // column-space split for occupancy
#define TILES_PER_SLICE (N_ROWS / 16 / SLICES)   // 128 j-tiles (2048 columns) per slice
#define COLS_PER_SLICE (TILES_PER_SLICE * 16)

// ---------------- Kernel 1: L2-normalize rows (f = x / max(||x||, 1e-12)) ----------------
__global__ __launch_bounds__(256) void supcon_normalize(const float* __restrict__ feat,
                                                        float* __restrict__ fn) {
    const int row = blockIdx.x;
    const int tid = threadIdx.x;                  // 256 threads == DIM
    const float x = feat[(size_t)row * DIM + tid];
    float s = x * x;
    #pragma unroll
    for (int m = 16; m >= 1; m >>= 1) s += __shfl_xor(s, m, 32);
    __shared__ float red[8];
    if ((tid & 31) == 0) red[tid >> 5] = s;
    __syncthreads();
    float tot = 0.f;
    #pragma unroll
    for (int i = 0; i < 8; ++i) tot += red[i];
    const float inv = 1.0f / fmaxf(sqrtf(tot), 1e-12f);
    fn[(size_t)row * DIM + tid] = x * inv;
}

// ---------------- Kernel 2: fused tiled f fᵀ + streaming softmax statistics ----------------
// Grid: (N/64)*SLICES workgroups of 128 threads (4 waves). Wave w owns row-tile rowBase+16w;
// each WG covers a 1/SLICES slice of the columns and emits partial Z/P/C for its 64 rows.
__global__ __launch_bounds__(128) void supcon_tiles(const float* __restrict__ fn,
                                                    const int*   __restrict__ labels,
                                                    float* __restrict__ Zp,
                                                    float* __restrict__ Pp,
                                                    float* __restrict__ Cp) {
    __shared__ float ldsA[64 * LDS_STRIDE];
    __shared__ float ldsB[16 * LDS_STRIDE];
    __shared__ int   ldsLab[COLS_PER_SLICE];

    const int tid      = threadIdx.x;
    const int lane     = tid & 31;
    const int wave     = tid >> 5;
    const int lane16   = lane & 15;
    const int halfSel  = lane >> 4;           // 0: lanes 0-15 (rows 0-7), 1: lanes 16-31 (rows 8-15)
    const int kOff     = halfSel * 2;         // k-pair offset within each 4-wide WMMA step
    const int slice    = blockIdx.x & (SLICES - 1);
    const int rowBase  = (blockIdx.x >> 2) * 64;
    const int colStart = slice * COLS_PER_SLICE;
    const int myTileRow = wave * 16;
    const int rowTag   = rowBase + myTileRow + halfSel * 8;  // first global row this lane owns

    // Loop-invariant per-thread prefetch offsets (8 x float4 per thread covers a 16x256 panel).
    int pfGlb[8];   // element offset within a B panel (row*DIM + col)
    int pfLds[8];   // element offset within ldsB       (row*LDS_STRIDE + col)
    #pragma unroll
    for (int u = 0; u < 8; ++u) {
        const int i  = tid + 128 * u;
        const int r  = i >> 6;
        const int cq = (i & 63) << 2;
        pfGlb[u] = r * DIM + cq;
        pfLds[u] = r * LDS_STRIDE + cq;
    }

    // Stage 64x256 A panel in LDS (float4 cooperative load, padded stride).
    for (int i = tid; i < 64 * 64; i += 128) {
        const int r  = i >> 6;
        const int cq = (i & 63) << 2;
        const float4 v = *(const float4*)(fn + (size_t)(rowBase + r) * DIM + cq);
        *(float4*)(&ldsA[r * LDS_STRIDE + cq]) = v;
    }
    // Stage this slice's column labels in LDS (8 KB).
    for (int i = tid; i < COLS_PER_SLICE; i += 128) ldsLab[i] = labels[colStart + i];
    // Stage B tile 0 of this slice.
    #pragma unroll
    for (int u = 0; u < 8; ++u)
        *(float4*)(&ldsB[pfLds[u]]) = *(const float4*)(fn + (size_t)colStart * DIM + pfGlb[u]);
    __syncthreads();

    // Per-lane labels for the 8 rows this lane accumulates (one-time global reads).
    int rowLab[8];
    #pragma unroll
    for (int v = 0; v < 8; ++v) rowLab[v] = labels[rowTag + v];

    float zAcc[8], pAcc[8], cAcc[8];
    #pragma unroll
    for (int v = 0; v < 8; ++v) { zAcc[v] = 0.f; pAcc[v] = 0.f; cAcc[v] = 0.f; }

    const int aRowOff = (myTileRow + lane16) * LDS_STRIDE;
    const int bRowOff = lane16 * LDS_STRIDE;
    const float* const slicePanels = fn + (size_t)colStart * DIM;   // base of this slice's B data

    for (int jt = 0; jt < TILES_PER_SLICE; ++jt) {
        // Issue next tile's global loads NOW; commit to LDS after the post-compute barrier.
        const int jn = (jt + 1 < TILES_PER_SLICE) ? (jt + 1) : jt;  // last iter: redundant reload
        const float* const nxtPanel = slicePanels + (size_t)jn * 16 * DIM;  // uniform (SALU) part
        float4 nxt[8];
        #pragma unroll
        for (int u = 0; u < 8; ++u) nxt[u] = *(const float4*)(nxtPanel + pfGlb[u]);

        // Column label for this lane's column (LDS, issued before the WMMA loop).
        const int colLab = ldsLab[jt * 16 + lane16];

        // 16x16 tile of f f^T via 64 chained V_WMMA_F32_16X16X4_F32 (exact f32 math).
        v8f acc = {0.f, 0.f, 0.f, 0.f, 0.f, 0.f, 0.f, 0.f};
        #pragma unroll 8
        for (int k = 0; k < DIM; k += 4) {
            v2f a = *(const v2f*)(&ldsA[aRowOff + k + kOff]);
            v2f b = *(const v2f*)(&ldsB[bRowOff + k + kOff]);
            acc = __builtin_amdgcn_wmma_f32_16x16x4_f32(
                /*neg_a=*/false, a, /*neg_b=*/false, b,
                /*c_mod=*/(short)0, acc, /*reuse_a=*/false, /*reuse_b=*/false);
        }

        // Branchless streaming epilogue. acc[v] = sim(row rowTag+v, col gCol).
        const int gCol = colStart + jt * 16 + lane16;
        #pragma unroll
        for (int v = 0; v < 8; ++v) {
            const float l  = acc[v] * INV_TEMP;
            const float ns = (rowTag + v != gCol) ? 1.0f : 0.0f;   // not-self mask
            const float mt = (rowLab[v] == colLab) ? 1.0f : 0.0f;  // label match
            const float mm = ns * mt;
            zAcc[v] = fmaf(ns, __expf(l), zAcc[v]);
            pAcc[v] = fmaf(mm, l, pAcc[v]);
            cAcc[v] += mm;
        }

        __syncthreads();                       // all waves done reading ldsB
        #pragma unroll
        for (int u = 0; u < 8; ++u) *(float4*)(&ldsB[pfLds[u]]) = nxt[u];
        __syncthreads();                       // next tile visible
    }

    // Reduce the 16 column-lanes sharing each row; lanes 0 and 16 own the results.
    #pragma unroll
    for (int v = 0; v < 8; ++v) {
        float z = zAcc[v], p = pAcc[v], c = cAcc[v];
        #pragma unroll
        for (int m = 8; m >= 1; m >>= 1) {
            z += __shfl_xor(z, m, 32);
            p += __shfl_xor(p, m, 32);
            c += __shfl_xor(c, m, 32);
        }
        if (lane16 == 0) {
            const int idx = slice * N_ROWS + rowTag + v;   // per-slice partials
            Zp[idx] = z; Pp[idx] = p; Cp[idx] = c;
        }
    }
}

// ---------------- Kernel 3: combine slices + scalar loss reduction ----------------
__global__ __launch_bounds__(256) void supcon_finalize(const float* __restrict__ Zp,
                                                       const float* __restrict__ Pp,
                                                       const float* __restrict__ Cp,
                                                       float* __restrict__ out) {
    __shared__ float red[256];
    float s = 0.f;
    for (int i = threadIdx.x; i < N_ROWS; i += 256) {
        float z = 0.f, p = 0.f, c = 0.f;
        #pragma unroll
        for (int sl = 0; sl < SLICES; ++sl) {
            z += Zp[sl * N_ROWS + i];
            p += Pp[sl * N_ROWS + i];
            c += Cp[sl * N_ROWS + i];
        }
        s += (p - c * logf(z + 1e-6f)) / fmaxf(c, 1.0f);
    }
    red[threadIdx.x] = s;
    __syncthreads();
    for (int off = 128; off > 0; off >>= 1) {
        if (threadIdx.x < off) red[threadIdx.x] += red[threadIdx.x + off];
        __syncthreads();
    }
    if (threadIdx.x == 0) out[0] = -red[0] / (float)N_ROWS;
}

extern "C" void kernel_launch(void* const* d_in, const int* in_sizes, int n_in,
                              void* d_out, int out_size, void* d_ws, size_t ws_size,
                              hipStream_t stream) {
    const float* feat   = (const float*)d_in[0];
    const int*   labels = (const int*)d_in[1];

    float* fn = (float*)d_ws;                        // 8192*256 f32 = 8 MB
    float* Zp = fn + (size_t)N_ROWS * DIM;           // SLICES * N_ROWS each
    float* Pp = Zp + (size_t)SLICES * N_ROWS;
    float* Cp = Pp + (size_t)SLICES * N_ROWS;

    supcon_normalize<<<N_ROWS, 256, 0, stream>>>(feat, fn);
    supcon_tiles<<<(N_ROWS / 64) * SLICES, 128, 0, stream>>>(fn, labels, Zp, Pp, Cp);
    supcon_finalize<<<1, 256, 0, stream>>>(Zp, Pp, Cp, (float*)d_out);
}